// WeightedLinearProbing_54820962566693
// MI455X (gfx1250) — compile-verified
//
#include <hip/hip_runtime.h>

// ---------------------------------------------------------------------------
// WeightedLinearProbing on MI455X (gfx1250, wave32).
// Streaming passes over x (512MB) + y (100MB) dominate; everything is built
// to touch them once with coalesced loads and zero floating-point atomics.
// The small [1000,256]x[256,50] GEMM uses V_WMMA_F32_16X16X4_F32 with
// clamped (not predicated) tile loads so the K-loop is load/load/wmma only.
// ---------------------------------------------------------------------------

constexpr int D   = 256;   // feature dim
constexpr int YD  = 50;    // target dim
constexpr int L   = 1000;  // num labels
constexpr int ZLD = 64;    // padded leading dim for logits z
constexpr int CS  = 512;   // rows per binning chunk
constexpr int CPT_MAX = 16;

typedef float v2f __attribute__((ext_vector_type(2)));
typedef float v8f __attribute__((ext_vector_type(8)));

// --------------------------- 1) per-chunk histogram -------------------------
__global__ void k_hist(const int* __restrict__ labels, int n, int nchunks,
                       int* __restrict__ hist /* [L][nchunks] */) {
  __shared__ int h[L];
  const int c = blockIdx.x;
  for (int i = threadIdx.x; i < L; i += blockDim.x) h[i] = 0;
  __syncthreads();
  const int row = c * CS + threadIdx.x;
  if (row < n) atomicAdd(&h[labels[row]], 1);  // integer: order-independent
  __syncthreads();
  for (int i = threadIdx.x; i < L; i += blockDim.x)
    hist[(long long)i * nchunks + c] = h[i];
}

// --------------------------- 2) counts per label ----------------------------
__global__ void k_counts(const int* __restrict__ hist, int nchunks,
                         int* __restrict__ counts) {
  __shared__ int s[256];
  const int l = blockIdx.x;
  int acc = 0;
  for (int c = threadIdx.x; c < nchunks; c += blockDim.x)
    acc += hist[(long long)l * nchunks + c];
  s[threadIdx.x] = acc;
  __syncthreads();
  for (int off = 128; off > 0; off >>= 1) {
    if (threadIdx.x < off) s[threadIdx.x] += s[threadIdx.x + off];
    __syncthreads();
  }
  if (threadIdx.x == 0) counts[l] = s[0];
}

// ------------------ 3) exclusive scan of counts -> offsets ------------------
__global__ void k_offsets(const int* __restrict__ counts,
                          int* __restrict__ offsets) {
  __shared__ int s[256];
  const int t = threadIdx.x;
  constexpr int CPT = (L + 255) / 256;  // 4
  int vals[CPT];
  int tot = 0;
#pragma unroll
  for (int i = 0; i < CPT; ++i) {
    int idx = t * CPT + i;
    vals[i] = (idx < L) ? counts[idx] : 0;
    tot += vals[i];
  }
  s[t] = tot;
  __syncthreads();
  for (int off = 1; off < 256; off <<= 1) {
    int v = (t >= off) ? s[t - off] : 0;
    __syncthreads();
    s[t] += v;
    __syncthreads();
  }
  int base = s[t] - tot;  // exclusive prefix across threads
#pragma unroll
  for (int i = 0; i < CPT; ++i) {
    int idx = t * CPT + i;
    if (idx < L) offsets[idx] = base;
    base += vals[i];
  }
}

// ----- 4) per-label scan over chunks: hist -> chunk base cursors (in-place) -
__global__ void k_chunkscan(int* __restrict__ hist, int nchunks,
                            const int* __restrict__ offsets) {
  __shared__ int s[256];
  const int l = blockIdx.x;
  const int t = threadIdx.x;
  const int cpt = (nchunks + 255) / 256;  // 4 for N=500000
  int vals[CPT_MAX];
  int tot = 0;
  for (int i = 0; i < cpt; ++i) {
    int c = t * cpt + i;
    vals[i] = (c < nchunks) ? hist[(long long)l * nchunks + c] : 0;
    tot += vals[i];
  }
  s[t] = tot;
  __syncthreads();
  for (int off = 1; off < 256; off <<= 1) {
    int v = (t >= off) ? s[t - off] : 0;
    __syncthreads();
    s[t] += v;
    __syncthreads();
  }
  int base = offsets[l] + s[t] - tot;
  for (int i = 0; i < cpt; ++i) {
    int c = t * cpt + i;
    if (c < nchunks) hist[(long long)l * nchunks + c] = base;
    base += vals[i];
  }
}

// ---------- 5) stable scatter: rank-within-chunk, fully deterministic -------
__global__ void k_scatter(const int* __restrict__ labels, int n, int nchunks,
                          const int* __restrict__ cb /* chunk bases */,
                          int* __restrict__ bucket) {
  __shared__ int sl[CS];
  const int c = blockIdx.x;
  const int t = threadIdx.x;
  const int row = c * CS + t;
  const int lab = (row < n) ? labels[row] : -1;
  sl[t] = lab;
  __syncthreads();
  if (lab >= 0) {
    int rank = 0;
    for (int j = 0; j < t; ++j) rank += (sl[j] == lab);
    bucket[cb[(long long)lab * nchunks + c] + rank] = row;
  }
}

// ------ 6) the big pass: gate = sigmoid(x.w1+b1); segment-mean of gate*x ----
// One block per label, thread j owns column j. Rows consumed in bucket order
// (ascending global row index) -> deterministic accumulation order.
__global__ void k_segreduce(const float* __restrict__ x,
                            const float* __restrict__ y,
                            const float* __restrict__ w1,
                            const float* __restrict__ b1,
                            const int* __restrict__ counts,
                            const int* __restrict__ offsets,
                            const int* __restrict__ bucket,
                            float* __restrict__ res,
                            float* __restrict__ ymean) {
  constexpr int RPI = 4;  // rows per barrier pair
  const int l    = blockIdx.x;
  const int j    = threadIdx.x;  // 0..255 == column
  const int wid  = j >> 5;
  const int lane = j & 31;
  __shared__ float s_part[RPI][8];
  __shared__ float s_gate[RPI];
  const float w1j = w1[j];
  const float b1v = b1[0];
  const int start = offsets[l];
  const int cnt   = counts[l];
  const int end   = start + cnt;
  float accx = 0.f;
  float accy = 0.f;  // meaningful only for j < YD
  for (int r0 = start; r0 < end; r0 += RPI) {
    float xv[RPI];
    int   rows[RPI];
    float valid[RPI];
#pragma unroll
    for (int t = 0; t < RPI; ++t) {
      int  r  = r0 + t;
      bool v  = (r < end);
      int  rr = v ? r : (end - 1);
      rows[t]  = bucket[rr];
      valid[t] = v ? 1.f : 0.f;
      xv[t] = x[(long long)rows[t] * D + j];
      float p = xv[t] * w1j;
#pragma unroll
      for (int off = 16; off > 0; off >>= 1) p += __shfl_down(p, off, 32);
      if (lane == 0) s_part[t][wid] = p;
    }
    __syncthreads();
    if (j < RPI) {
      float s = 0.f;
#pragma unroll
      for (int w = 0; w < 8; ++w) s += s_part[j][w];
      s_gate[j] = 1.f / (1.f + __expf(-(s + b1v)));
    }
    __syncthreads();
#pragma unroll
    for (int t = 0; t < RPI; ++t) {
      float g = s_gate[t] * valid[t];
      accx += g * xv[t];
      if (j < YD) accy += valid[t] * y[(long long)rows[t] * YD + j];
    }
  }
  const float denom = fmaxf((float)cnt, 1.f);
  res[l * D + j] = accx / denom;
  if (j < YD) ymean[l * YD + j] = accy / denom;
}

// --------- 7) z = res @ w2 + b2 via V_WMMA_F32_16X16X4_F32 (wave32) ---------
// A: 16x4 f32 per lane = 2 VGPRs (VGPR v, half g -> K = v + 2g, M = lane&15)
// B: 4x16 f32          = 2 VGPRs (VGPR v, half g -> K = v + 2g, N = lane&15)
// D: 16x16 f32         = 8 VGPRs (VGPR v, half g -> M = v + 8g, N = lane&15)
// Out-of-range lanes (m>=L or n>=YD) only influence output elements that are
// never stored, so their addresses are CLAMPED (no per-iteration predication,
// no EXEC churn in the K-loop; EXEC stays all-ones for the WMMAs).
__global__ void k_gemm_wmma(const float* __restrict__ res,
                            const float* __restrict__ w2,
                            const float* __restrict__ b2,
                            float* __restrict__ z /* [L][ZLD] */) {
  const int lane = threadIdx.x & 31;
  const int wave = blockIdx.x * (blockDim.x >> 5) + (threadIdx.x >> 5);
  constexpr int TILES_M = (L + 15) / 16;   // 63
  constexpr int TILES_N = (YD + 15) / 16;  // 4
  if (wave >= TILES_M * TILES_N) return;   // whole-wave uniform exit
  const int tm   = wave / TILES_N;
  const int tn   = wave % TILES_N;
  const int half = lane >> 4;   // 0 or 1
  const int l16  = lane & 15;
  const int m  = tm * 16 + l16;            // logical A row for this lane
  const int n  = tn * 16 + l16;            // logical B col for this lane
  const int mc = (m < L)  ? m : (L - 1);   // clamped load addresses
  const int nc = (n < YD) ? n : (YD - 1);
  const float* __restrict__ arow = res + (long long)mc * D + 2 * half;
  const float* __restrict__ bcol = w2 + nc + (long long)(2 * half) * YD;
  v8f acc = {};
  for (int kb = 0; kb < D; kb += 4) {
    v2f a = *(const v2f*)(arow + kb);      // K = kb + 2*half + {0,1}, contiguous
    v2f b;
    b[0] = bcol[(long long)kb * YD];       // K = kb + 2*half
    b[1] = bcol[(long long)kb * YD + YD];  // K = kb + 2*half + 1
    acc = __builtin_amdgcn_wmma_f32_16x16x4_f32(
        /*neg_a=*/false, a, /*neg_b=*/false, b,
        /*c_mod=*/(short)0, acc, /*reuse_a=*/false, /*reuse_b=*/false);
  }
  const float bias = b2[nc];
#pragma unroll
  for (int v = 0; v < 8; ++v) {
    int mm = tm * 16 + v + 8 * half;
    if (mm < L && n < YD) z[mm * ZLD + n] = acc[v] + bias;
  }
}

// ------- 8) per-row: softmax -> log_softmax -> dot(y_mean) -> rowloss -------
__global__ void k_loss_rows(const float* __restrict__ z,
                            const float* __restrict__ ymean,
                            float* __restrict__ rowloss) {
  const int lane = threadIdx.x & 31;
  const int row  = blockIdx.x * (blockDim.x >> 5) + (threadIdx.x >> 5);
  if (row >= L) return;
  const int  j0 = lane;        // always < YD (YD=50 > 32)
  const int  j1 = lane + 32;
  const bool v1 = (j1 < YD);
  const float NEG_INF = -__builtin_inff();
  float z0 = z[row * ZLD + j0];
  float z1 = v1 ? z[row * ZLD + j1] : NEG_INF;
  float m = fmaxf(z0, z1);
#pragma unroll
  for (int off = 16; off > 0; off >>= 1) m = fmaxf(m, __shfl_xor(m, off, 32));
  float e0 = __expf(z0 - m);
  float e1 = v1 ? __expf(z1 - m) : 0.f;
  float s = e0 + e1;
#pragma unroll
  for (int off = 16; off > 0; off >>= 1) s += __shfl_xor(s, off, 32);
  float p0 = e0 / s;
  float p1 = e1 / s;                    // 0 for invalid j1
  float s2 = __expf(p0) + (v1 ? __expf(p1) : 0.f);
#pragma unroll
  for (int off = 16; off > 0; off >>= 1) s2 += __shfl_xor(s2, off, 32);
  float lg = __logf(s2);
  float part = ymean[row * YD + j0] * (p0 - lg);
  if (v1) part += ymean[row * YD + j1] * (p1 - lg);
#pragma unroll
  for (int off = 16; off > 0; off >>= 1) part += __shfl_xor(part, off, 32);
  if (lane == 0) rowloss[row] = -part / (float)L;
}

// ------------------- 9) fixed-order final reduction -> d_out ----------------
__global__ void k_final(const float* __restrict__ rowloss, float* __restrict__ out) {
  __shared__ float s[256];
  float acc = 0.f;
  for (int i = threadIdx.x; i < L; i += 256) acc += rowloss[i];
  s[threadIdx.x] = acc;
  __syncthreads();
  for (int off = 128; off > 0; off >>= 1) {
    if (threadIdx.x < off) s[threadIdx.x] += s[threadIdx.x + off];
    __syncthreads();
  }
  if (threadIdx.x == 0) out[0] = s[0];
}

// ---------------------------------------------------------------------------
extern "C" void kernel_launch(void* const* d_in, const int* in_sizes, int n_in,
                              void* d_out, int out_size, void* d_ws, size_t ws_size,
                              hipStream_t stream) {
  const float* x      = (const float*)d_in[0];
  const int*   labels = (const int*)  d_in[1];
  const float* y      = (const float*)d_in[2];
  const float* w1     = (const float*)d_in[3];
  const float* b1     = (const float*)d_in[4];
  const float* w2     = (const float*)d_in[5];
  const float* b2     = (const float*)d_in[6];
  const int n = in_sizes[1];
  const int nchunks = (n + CS - 1) / CS;

  char* ws = (char*)d_ws;
  size_t off = 0;
  auto carve = [&](size_t bytes) -> void* {
    void* p = ws + off;
    off += (bytes + 255) & ~(size_t)255;
    return p;
  };
  int*   hist    = (int*)  carve((size_t)L * nchunks * sizeof(int)); // -> chunk bases
  int*   counts  = (int*)  carve((size_t)L * sizeof(int));
  int*   offsets = (int*)  carve((size_t)L * sizeof(int));
  int*   bucket  = (int*)  carve((size_t)n * sizeof(int));
  float* res     = (float*)carve((size_t)L * D * sizeof(float));
  float* ymean   = (float*)carve((size_t)L * YD * sizeof(float));
  float* z       = (float*)carve((size_t)L * ZLD * sizeof(float));
  float* rowloss = (float*)carve((size_t)L * sizeof(float));
  (void)ws_size; (void)n_in; (void)out_size;

  k_hist     <<<nchunks, CS,  0, stream>>>(labels, n, nchunks, hist);
  k_counts   <<<L,       256, 0, stream>>>(hist, nchunks, counts);
  k_offsets  <<<1,       256, 0, stream>>>(counts, offsets);
  k_chunkscan<<<L,       256, 0, stream>>>(hist, nchunks, offsets);
  k_scatter  <<<nchunks, CS,  0, stream>>>(labels, n, nchunks, hist, bucket);
  k_segreduce<<<L,       256, 0, stream>>>(x, y, w1, b1, counts, offsets, bucket,
                                           res, ymean);
  constexpr int TILES = ((L + 15) / 16) * ((YD + 15) / 16);  // 252 waves
  k_gemm_wmma<<<(TILES * 32 + 255) / 256, 256, 0, stream>>>(res, w2, b2, z);
  k_loss_rows<<<(L + 7) / 8, 256, 0, stream>>>(z, ymean, rowloss);
  k_final    <<<1, 256, 0, stream>>>(rowloss, (float*)d_out);
}